// LatentReasoningModule_89412629168276
// MI455X (gfx1250) — compile-verified
//
#include <hip/hip_runtime.h>

// ---------------------------------------------------------------------------
// LatentReasoningModule on MI455X (gfx1250), fp32 throughout.
// GEMMs: V_WMMA_F32_16X16X4_F32 with double-buffered LDS fed by
// GLOBAL_LOAD_ASYNC_TO_LDS_B128 (ASYNCcnt-tracked async copy engine).
// ---------------------------------------------------------------------------

typedef float v2f __attribute__((ext_vector_type(2)));
typedef float v8f __attribute__((ext_vector_type(8)));
typedef int   v4i __attribute__((vector_size(16)));   // matches builtin prototype

typedef __attribute__((address_space(1))) v4i* gv4i_p;   // global int4*
typedef __attribute__((address_space(3))) v4i* lv4i_p;   // LDS int4*

#define BM 128
#define BN 256
#define BK 32
#define LDSS 36   // padded LDS row stride: 36*4B = 144B -> 16B-aligned rows, bank-spread

// One 16-byte global->LDS async copy per lane (ASYNCcnt++ on issue).
__device__ __forceinline__ void async_copy_b128(const float* g, float* l) {
    __builtin_amdgcn_global_load_async_to_lds_b128(
        (gv4i_p)(g), (lv4i_p)(l), /*offset=*/0, /*cpol=*/0);
}

// Y[M,N] = act(X[M,K] @ W[N,K]^T + bias[N]),  act = relu if RELU.
// 256 threads = 8 waves (2x4); block tile 128x256; wave tile 64x64.
template<bool RELU>
__global__ __launch_bounds__(256)
void gemm_bias_kernel(const float* __restrict__ X, const float* __restrict__ W,
                      const float* __restrict__ bias, float* __restrict__ Y,
                      int M, int N, int K)
{
    __shared__ float sA[2][BM * LDSS];   // 2 x 128x32 (padded)
    __shared__ float sB[2][BN * LDSS];   // 2 x 256x32 (padded)

    const int tid   = threadIdx.x;
    const int lane  = tid & 31;
    const int wave  = tid >> 5;      // 0..7
    const int waveM = wave >> 2;     // 0..1
    const int waveN = wave & 3;      // 0..3
    const int khalf = lane >> 4;     // 0 or 1
    const int l16   = lane & 15;

    const int tileM = blockIdx.y * BM;
    const int tileN = blockIdx.x * BN;

    v8f acc[4][4] = {};

    // --- async stage of one K-block into LDS buffer `buf` (12 copies/lane) ---
    auto stage = [&](int buf, int k0) {
        #pragma unroll
        for (int it = 0; it < 4; ++it) {              // A tile: 128x32 = 1024 chunks
            int c   = tid + it * 256;
            int row = c >> 3;
            int c4  = (c & 7) * 4;
            async_copy_b128(&X[(size_t)(tileM + row) * K + k0 + c4],
                            &sA[buf][row * LDSS + c4]);
        }
        #pragma unroll
        for (int it = 0; it < 8; ++it) {              // B tile: 256x32 = 2048 chunks
            int c   = tid + it * 256;
            int row = c >> 3;
            int c4  = (c & 7) * 4;
            async_copy_b128(&W[(size_t)(tileN + row) * K + k0 + c4],
                            &sB[buf][row * LDSS + c4]);
        }
    };

    const int nk = K / BK;
    stage(0, 0);

    for (int i = 0; i < nk; ++i) {
        const int cur = i & 1;
        if (i + 1 < nk) {
            stage(1 - cur, (i + 1) * BK);
            // 12 copies for buffer (1-cur) may still be in flight; async loads
            // retire in order, so <=12 outstanding means `cur` has landed.
            __builtin_amdgcn_s_wait_asynccnt(12);
        } else {
            __builtin_amdgcn_s_wait_asynccnt(0);
        }
        __syncthreads();   // all waves' portions of `cur` visible in LDS

        #pragma unroll
        for (int kk = 0; kk < BK; kk += 4) {
            // A frag (16x4 fp32): lanes 0-15 K={kk,kk+1}, lanes 16-31 K={kk+2,kk+3}
            v2f a[4], b[4];
            #pragma unroll
            for (int mf = 0; mf < 4; ++mf) {
                int row = waveM * 64 + mf * 16 + l16;
                a[mf] = *reinterpret_cast<const v2f*>(
                    &sA[cur][row * LDSS + kk + 2 * khalf]);
            }
            // B frag (4x16): B[k][n] = W[n][k]; same K split, N = lane&15
            #pragma unroll
            for (int nf = 0; nf < 4; ++nf) {
                int col = waveN * 64 + nf * 16 + l16;
                b[nf] = *reinterpret_cast<const v2f*>(
                    &sB[cur][col * LDSS + kk + 2 * khalf]);
            }
            #pragma unroll
            for (int mf = 0; mf < 4; ++mf)
                #pragma unroll
                for (int nf = 0; nf < 4; ++nf)
                    acc[mf][nf] = __builtin_amdgcn_wmma_f32_16x16x4_f32(
                        false, a[mf], false, b[nf],
                        (short)0, acc[mf][nf], false, false);
        }
        __syncthreads();   // all waves done reading `cur` before it is re-staged
    }

    // Epilogue: C/D layout -> row = r + 8*khalf, col = lane&15
    #pragma unroll
    for (int mf = 0; mf < 4; ++mf) {
        #pragma unroll
        for (int nf = 0; nf < 4; ++nf) {
            int col = tileN + waveN * 64 + nf * 16 + l16;
            float bv = bias[col];
            #pragma unroll
            for (int r = 0; r < 8; ++r) {
                int row = tileM + waveM * 64 + mf * 16 + r + 8 * khalf;
                float v = acc[mf][nf][r] + bv;
                if (RELU) v = fmaxf(v, 0.0f);
                Y[(size_t)row * N + col] = v;
            }
        }
    }
}

// ---------------------------------------------------------------------------
// GRU gate math (PyTorch GRUCell semantics), x == h == latent.
// ---------------------------------------------------------------------------
__device__ __forceinline__ float sigmoid_fast(float x) {
    return 1.0f / (1.0f + __expf(-x));
}
__device__ __forceinline__ float tanh_fast(float x) {
    return 1.0f - 2.0f / (__expf(2.0f * x) + 1.0f);
}
__device__ __forceinline__ float gru_one(float ir, float iz, float in_,
                                         float hr, float hz, float hn, float h) {
    float r = sigmoid_fast(ir + hr);
    float z = sigmoid_fast(iz + hz);
    float n = tanh_fast(in_ + r * hn);
    return (1.0f - z) * n + z * h;
}

// lat: [M,512] updated in place; gi/gh: [M,1536]; traj: [M, steps, 512]
__global__ __launch_bounds__(256)
void gru_update_kernel(float* __restrict__ lat, const float* __restrict__ gi,
                       const float* __restrict__ gh, float* __restrict__ traj,
                       int step, int steps)
{
    int idx = blockIdx.x * 256 + threadIdx.x;   // over M * (512/4)
    int m = idx >> 7;
    int l = (idx & 127) * 4;
    size_t gb = (size_t)m * 1536 + l;

    float4 ir  = *reinterpret_cast<const float4*>(gi + gb);
    float4 iz  = *reinterpret_cast<const float4*>(gi + gb + 512);
    float4 inn = *reinterpret_cast<const float4*>(gi + gb + 1024);
    float4 hr  = *reinterpret_cast<const float4*>(gh + gb);
    float4 hz  = *reinterpret_cast<const float4*>(gh + gb + 512);
    float4 hn  = *reinterpret_cast<const float4*>(gh + gb + 1024);
    float4 h   = *reinterpret_cast<const float4*>(lat + (size_t)m * 512 + l);

    float4 o;
    o.x = gru_one(ir.x, iz.x, inn.x, hr.x, hz.x, hn.x, h.x);
    o.y = gru_one(ir.y, iz.y, inn.y, hr.y, hz.y, hn.y, h.y);
    o.z = gru_one(ir.z, iz.z, inn.z, hr.z, hz.z, hn.z, h.z);
    o.w = gru_one(ir.w, iz.w, inn.w, hr.w, hz.w, hn.w, h.w);

    *reinterpret_cast<float4*>(lat + (size_t)m * 512 + l) = o;
    *reinterpret_cast<float4*>(traj + ((size_t)m * steps + step) * 512 + l) = o;
}

// ---------------------------------------------------------------------------
extern "C" void kernel_launch(void* const* d_in, const int* in_sizes, int n_in,
                              void* d_out, int out_size, void* d_ws, size_t ws_size,
                              hipStream_t stream)
{
    const int M = 8192;   // B*S = 4*2048
    const int H = 2048;
    const int L = 512;

    const float* X      = (const float*)d_in[0];
    // d_in[1] = num_reasoning_steps (device scalar; unreadable during capture).
    const float* W_enc1 = (const float*)d_in[2];
    const float* b_enc1 = (const float*)d_in[3];
    const float* W_enc2 = (const float*)d_in[4];
    const float* b_enc2 = (const float*)d_in[5];
    const float* W_dec1 = (const float*)d_in[6];
    const float* b_dec1 = (const float*)d_in[7];
    const float* W_dec2 = (const float*)d_in[8];
    const float* b_dec2 = (const float*)d_in[9];
    const float* w_ih   = (const float*)d_in[10];
    const float* w_hh   = (const float*)d_in[11];
    const float* b_ih   = (const float*)d_in[12];
    const float* b_hh   = (const float*)d_in[13];

    float* out  = (float*)d_out;                     // decoded [M,H]
    float* traj = out + (size_t)M * H;               // trajectory [M, steps, L]
    // Derive step count from the fixed output buffer size (deterministic,
    // capture-safe): out_size = M*H + M*steps*L.
    int steps = (int)(((size_t)out_size - (size_t)M * H) / ((size_t)M * L));
    if (steps < 1) steps = 1;

    // Workspace layout (fp32): lat[M,L] | tmp1[M,1024] | gi[M,1536] | gh[M,1536]
    float* lat  = (float*)d_ws;
    float* tmp1 = lat  + (size_t)M * L;
    float* gi   = tmp1 + (size_t)M * 1024;
    float* gh   = gi   + (size_t)M * 1536;

    dim3 blk(256);

    // encode: tmp1 = relu(X @ W_enc1^T + b_enc1)        [M,1024], K=2048
    gemm_bias_kernel<true ><<<dim3(1024 / BN, M / BM), blk, 0, stream>>>(
        X, W_enc1, b_enc1, tmp1, M, 1024, 2048);
    // lat = tmp1 @ W_enc2^T + b_enc2                    [M,512],  K=1024
    gemm_bias_kernel<false><<<dim3(512 / BN, M / BM), blk, 0, stream>>>(
        tmp1, W_enc2, b_enc2, lat, M, 512, 1024);

    // GRU reasoning steps
    for (int s = 0; s < steps; ++s) {
        gemm_bias_kernel<false><<<dim3(1536 / BN, M / BM), blk, 0, stream>>>(
            lat, w_ih, b_ih, gi, M, 1536, 512);
        gemm_bias_kernel<false><<<dim3(1536 / BN, M / BM), blk, 0, stream>>>(
            lat, w_hh, b_hh, gh, M, 1536, 512);
        gru_update_kernel<<<dim3((M * (L / 4)) / 256), blk, 0, stream>>>(
            lat, gi, gh, traj, s, steps);
    }

    // decode: tmp1 = relu(lat @ W_dec1^T + b_dec1)      [M,1024], K=512
    gemm_bias_kernel<true ><<<dim3(1024 / BN, M / BM), blk, 0, stream>>>(
        lat, W_dec1, b_dec1, tmp1, M, 1024, 512);
    // out = tmp1 @ W_dec2^T + b_dec2                    [M,2048], K=1024
    gemm_bias_kernel<false><<<dim3(2048 / BN, M / BM), blk, 0, stream>>>(
        tmp1, W_dec2, b_dec2, out, M, 2048, 1024);
}